// KeypointRenderer_6047313952775
// MI455X (gfx1250) — compile-verified
//
#include <hip/hip_runtime.h>
#include <hip/hip_bf16.h>

typedef __attribute__((ext_vector_type(2))) float v2f;
typedef __attribute__((ext_vector_type(8))) float v8f;

#define HW        128
#define K_KP      21
#define INV_S2    0.25f     //  1/sigma^2,  sigma = 2
#define NH_INV_S2 -0.125f   // -1/(2*sigma^2)
#define BIG       1e30f

static __device__ __forceinline__ float max16(const v8f& d0, const v8f& d1) {
    float s = d0[0];
#pragma unroll
    for (int i = 1; i < 8; ++i) s = fmaxf(s, d0[i]);
#pragma unroll
    for (int i = 0; i < 8; ++i) s = fmaxf(s, d1[i]);
    return s;
}

__global__ __launch_bounds__(256)
void KeypointRenderer_kernel(const float* __restrict__ kp,     // [256,21,3]
                             const float* __restrict__ boxes,  // [256,4]
                             float* __restrict__ out)          // [256,1,128,128]
{
    const int box  = blockIdx.x;
    const int lane = threadIdx.x & 31;
    const int wave = threadIdx.x >> 5;
    const int n    = lane & 15;          // pixel-in-tile index AND keypoint index (mod 16)
    const bool lo  = lane < 16;

    // Box transform (uniform within block)
    const float x0 = boxes[box * 4 + 0];
    const float y0 = boxes[box * 4 + 1];
    const float bw = fmaxf(boxes[box * 4 + 2] - x0, 1e-6f);
    const float bh = fmaxf(boxes[box * 4 + 3] - y0, 1e-6f);
    const float sx = 128.0f / bw;
    const float sy = 128.0f / bh;

    // ---- A matrices (16x4 f32, 2 VGPRs) for keypoints 0-15 and 16-31 ----
    // Row m = [kx, ky, kx^2+ky^2, 1]; invisible/padded rows get ss=BIG -> exp -> 0.
    // Layout: lanes 0-15 hold K-elems {0,1}, lanes 16-31 hold {2,3}, M = lane%16.
    v2f a_t0, a_t1;
    {
        const float* p = kp + (box * K_KP + n) * 3;   // keypoint n (tile 0)
        const bool  v  = p[2] > 0.1f;
        const float kx = v ? (p[0] - x0) * sx : 0.0f;
        const float ky = v ? (p[1] - y0) * sy : 0.0f;
        const float ss = v ? (kx * kx + ky * ky) : BIG;
        a_t0[0] = lo ? kx : ss;
        a_t0[1] = lo ? ky : 1.0f;
    }
    {
        const int   m  = 16 + n;                      // keypoint m (tile 1), pad m>=21
        const int   mc = (m < K_KP) ? m : (K_KP - 1);
        const float* p = kp + (box * K_KP + mc) * 3;
        const bool  v  = (m < K_KP) && (p[2] > 0.1f);
        const float kx = v ? (p[0] - x0) * sx : 0.0f;
        const float ky = v ? (p[1] - y0) * sy : 0.0f;
        const float ss = v ? (kx * kx + ky * ky) : BIG;
        a_t1[0] = lo ? kx : ss;
        a_t1[1] = lo ? ky : 1.0f;
    }

    float* obase = out + (size_t)box * (HW * HW);

    // Each wave owns 16 rows; each row = 4 pixel-tile pairs (32 px per iteration).
    const int r0 = wave * 16;
    for (int row = r0; row < r0 + 16; ++row) {
        const float fy    = (float)row;
        const float by_lo = INV_S2 * fy;      // hoisted row terms
        const float fy2   = fy * fy;
        float* orow = obase + row * HW;

#pragma unroll
        for (int xb = 0; xb < HW; xb += 32) {
            const float fxA = (float)(xb + n);
            const float fxB = fxA + 16.0f;

            // B (4x16): col = [x/s2, y/s2, -1/(2s2), -(x^2+y^2)/(2s2)]
            // lanes 0-15 hold rows {0,1}; lanes 16-31 rows {2,3}; N = lane%16.
            v2f bA, bB;
            bA[0] = lo ? (INV_S2 * fxA) : NH_INV_S2;
            bA[1] = lo ? by_lo          : (NH_INV_S2 * (fxA * fxA + fy2));
            bB[0] = lo ? (INV_S2 * fxB) : NH_INV_S2;
            bB[1] = lo ? by_lo          : (NH_INV_S2 * (fxB * fxB + fy2));

            v8f c = {};
            // s(m,n) = -d2/(2*sigma^2): 32 keypoints x 32 pixels via 4 WMMAs
            v8f dA0 = __builtin_amdgcn_wmma_f32_16x16x4_f32(
                          false, a_t0, false, bA, (short)0, c, false, false);
            v8f dA1 = __builtin_amdgcn_wmma_f32_16x16x4_f32(
                          false, a_t1, false, bA, (short)0, c, false, false);
            v8f dB0 = __builtin_amdgcn_wmma_f32_16x16x4_f32(
                          false, a_t0, false, bB, (short)0, c, false, false);
            v8f dB1 = __builtin_amdgcn_wmma_f32_16x16x4_f32(
                          false, a_t1, false, bB, (short)0, c, false, false);

            // log-domain max over this lane's keypoint half for each tile
            const float sA = max16(dA0, dA1);
            const float sB = max16(dB0, dB1);

            // Cross-half exchange: lane L ends with full max for tile-A pixel L,
            // lane L+16 with full max for tile-B pixel L.
            const float u = lo ? sA : sB;
            const float v = lo ? sB : sA;
            const float w = __shfl_xor(v, 16, 32);
            const float g = __expf(fmaxf(u, w));   // 1 transcendental per pixel

            // 32 lanes -> 32 consecutive floats: fully coalesced 128B store
            orow[xb + lane] = g;
        }
    }
}

extern "C" void kernel_launch(void* const* d_in, const int* in_sizes, int n_in,
                              void* d_out, int out_size, void* d_ws, size_t ws_size,
                              hipStream_t stream) {
    const float* kp    = (const float*)d_in[0];   // keypoints  [256,21,3] f32
    const float* boxes = (const float*)d_in[1];   // hand_boxes [256,4]    f32
    float* out = (float*)d_out;                   // [256,1,128,128] f32
    (void)in_sizes; (void)n_in; (void)out_size; (void)d_ws; (void)ws_size;

    KeypointRenderer_kernel<<<256, 256, 0, stream>>>(kp, boxes, out);
}